// GraphNudger_6906307412233
// MI455X (gfx1250) — compile-verified
//
#include <hip/hip_runtime.h>
#include <math.h>

#define ETA 0.01f
#define EPS 1e-8f

typedef __attribute__((ext_vector_type(2))) float v2f;
typedef __attribute__((ext_vector_type(4))) float v4f;
typedef __attribute__((ext_vector_type(8))) float v8f;

// ---------------------------------------------------------------------------
// Row L2 norms. One block per row. out[row] = inv ? 1/(||x||+eps) : ||x||
// ---------------------------------------------------------------------------
__global__ void rownorm_kernel(const float* __restrict__ X, int cols,
                               float* __restrict__ out, int inv) {
  __shared__ float red[256];
  const int row = blockIdx.x;
  const int tid = threadIdx.x;
  const float* rp = X + (size_t)row * cols;
  float s = 0.f;
  for (int c = tid; c < cols; c += 256) {
    float x = rp[c];
    s += x * x;
  }
  red[tid] = s;
  __syncthreads();
  for (int off = 128; off > 0; off >>= 1) {
    if (tid < off) red[tid] += red[tid + off];
    __syncthreads();
  }
  if (tid == 0) {
    float n = sqrtf(red[0]);
    out[row] = inv ? (1.0f / (n + EPS)) : n;
  }
}

// ---------------------------------------------------------------------------
// simT[s][b] = 0.5 + 0.5 * sinv[s] * hinv[b] * dot(Sign[s], Heat[b])
// One wave32 per 16x16 (s,b) tile, V_WMMA_F32_16X16X4_F32 over K=F.
//
// K-permutation trick: dot() is invariant to K order if A and B share the
// same lane->K mapping. Each half-wave's row pointer is offset by 4*half
// floats, so one aligned float4 covers the K-values of TWO consecutive WMMA
// fragments (.xy and .zw) -> B128 loads instead of strided B64 pairs.
// Software-pipelined one chunk deep so loads overlap the previous WMMAs.
// ---------------------------------------------------------------------------
__global__ void simgemm_kernel(const float* __restrict__ H,
                               const float* __restrict__ S,
                               const float* __restrict__ hinv,
                               const float* __restrict__ sinv,
                               float* __restrict__ simT,
                               int F, int Bn, int Srows) {
  const int wave   = (blockIdx.x * blockDim.x + threadIdx.x) >> 5;
  const int lane   = threadIdx.x & 31;
  const int bTiles = Bn >> 4;               // 4 for B=64
  const int sTile  = wave / bTiles;
  const int bTile  = wave % bTiles;
  if (sTile >= (Srows >> 4)) return;        // wave-uniform guard

  const int half = lane >> 4;               // 0 or 1
  const int mn   = lane & 15;

  // 4*half float offset: half 0 covers cols [k, k+3], half 1 covers [k+4, k+7]
  const float* sRow = S + (size_t)(sTile * 16 + mn) * F + 4 * half;
  const float* hRow = H + (size_t)(bTile * 16 + mn) * F + 4 * half;

  v8f c0 = {};
  v8f c1 = {};

  // prologue: first 16-col chunk
  v4f pa0 = *(const v4f*)(sRow + 0);
  v4f pb0 = *(const v4f*)(hRow + 0);
  v4f pa1 = *(const v4f*)(sRow + 8);
  v4f pb1 = *(const v4f*)(hRow + 8);

  for (int k = 16; k < F; k += 16) {
    // issue next chunk's loads before consuming current fragments
    v4f na0 = *(const v4f*)(sRow + k);
    v4f nb0 = *(const v4f*)(hRow + k);
    v4f na1 = *(const v4f*)(sRow + k + 8);
    v4f nb1 = *(const v4f*)(hRow + k + 8);

    c0 = __builtin_amdgcn_wmma_f32_16x16x4_f32(
        false, __builtin_shufflevector(pa0, pa0, 0, 1),
        false, __builtin_shufflevector(pb0, pb0, 0, 1), (short)0, c0, false, false);
    c1 = __builtin_amdgcn_wmma_f32_16x16x4_f32(
        false, __builtin_shufflevector(pa0, pa0, 2, 3),
        false, __builtin_shufflevector(pb0, pb0, 2, 3), (short)0, c1, false, false);
    c0 = __builtin_amdgcn_wmma_f32_16x16x4_f32(
        false, __builtin_shufflevector(pa1, pa1, 0, 1),
        false, __builtin_shufflevector(pb1, pb1, 0, 1), (short)0, c0, false, false);
    c1 = __builtin_amdgcn_wmma_f32_16x16x4_f32(
        false, __builtin_shufflevector(pa1, pa1, 2, 3),
        false, __builtin_shufflevector(pb1, pb1, 2, 3), (short)0, c1, false, false);

    pa0 = na0; pb0 = nb0; pa1 = na1; pb1 = nb1;
  }

  // epilogue: consume last chunk
  c0 = __builtin_amdgcn_wmma_f32_16x16x4_f32(
      false, __builtin_shufflevector(pa0, pa0, 0, 1),
      false, __builtin_shufflevector(pb0, pb0, 0, 1), (short)0, c0, false, false);
  c1 = __builtin_amdgcn_wmma_f32_16x16x4_f32(
      false, __builtin_shufflevector(pa0, pa0, 2, 3),
      false, __builtin_shufflevector(pb0, pb0, 2, 3), (short)0, c1, false, false);
  c0 = __builtin_amdgcn_wmma_f32_16x16x4_f32(
      false, __builtin_shufflevector(pa1, pa1, 0, 1),
      false, __builtin_shufflevector(pb1, pb1, 0, 1), (short)0, c0, false, false);
  c1 = __builtin_amdgcn_wmma_f32_16x16x4_f32(
      false, __builtin_shufflevector(pa1, pa1, 2, 3),
      false, __builtin_shufflevector(pb1, pb1, 2, 3), (short)0, c1, false, false);

  const int n  = mn;                        // C/D column = lane&15
  const float hv = 0.5f * hinv[bTile * 16 + n];
#pragma unroll
  for (int j = 0; j < 8; ++j) {
    const int m    = j + 8 * half;          // C/D row
    const int srow = sTile * 16 + m;
    const float acc = c0[j] + c1[j];
    simT[(size_t)srow * Bn + (bTile * 16 + n)] = 0.5f + hv * sinv[srow] * acc;
  }
}

// ---------------------------------------------------------------------------
// Zero scratch accumulator.
// ---------------------------------------------------------------------------
__global__ void zero_kernel(float* __restrict__ p, int n) {
  int i = blockIdx.x * blockDim.x + threadIdx.x;
  if (i < n) p[i] = 0.0f;
}

// ---------------------------------------------------------------------------
// Edge scatter. Each wave grabs a batch of 32 edges: every lane loads one
// edge's (sign, disease, weight) coalesced, then the wave iterates over the
// batch with __shfl broadcasts. Per edge: one contiguous 256B gather of
// simT[s][0..63] and 64 fp32 global atomics into agg[d][0..63].
// ---------------------------------------------------------------------------
__global__ void edge_scatter_kernel(const float* __restrict__ simT,
                                    const float* __restrict__ ew,
                                    const int* __restrict__ ed,
                                    const int* __restrict__ es,
                                    float* __restrict__ agg, int E) {
  const int wave   = (blockIdx.x * blockDim.x + threadIdx.x) >> 5;
  const int lane   = threadIdx.x & 31;
  const int nwaves = (gridDim.x * blockDim.x) >> 5;

  for (int base = wave * 32; base < E; base += nwaves * 32) {
    const int e = base + lane;
    int   s = 0, d = 0;
    float w = 0.0f;
    if (e < E) {                 // invalid lanes carry w=0 but are never used
      s = es[e];
      d = ed[e];
      w = ew[e];
    }
    const int cnt = (E - base < 32) ? (E - base) : 32;
    for (int i = 0; i < cnt; ++i) {
      const int   si = __shfl(s, i);
      const int   di = __shfl(d, i);
      const float wi = __shfl(w, i);
      const float* col = simT + (size_t)si * 64;
      float* dst = agg + (size_t)di * 64;
      atomicAdd(dst + lane,      wi * col[lane]);
      atomicAdd(dst + lane + 32, wi * col[lane + 32]);
    }
  }
}

// ---------------------------------------------------------------------------
// out[b][d] = ETA * gnorm[b] * agg[d][b]
// ---------------------------------------------------------------------------
__global__ void finalize_kernel(const float* __restrict__ agg,
                                const float* __restrict__ gnorm,
                                float* __restrict__ out, int D, int total) {
  int idx = blockIdx.x * blockDim.x + threadIdx.x;
  if (idx >= total) return;
  int b = idx / D;
  int d = idx - b * D;
  out[idx] = ETA * gnorm[b] * agg[(size_t)d * 64 + b];
}

// ---------------------------------------------------------------------------
extern "C" void kernel_launch(void* const* d_in, const int* in_sizes, int n_in,
                              void* d_out, int out_size, void* d_ws, size_t ws_size,
                              hipStream_t stream) {
  (void)n_in; (void)ws_size;
  const float* heat = (const float*)d_in[0];  // [B,F]
  const float* grad = (const float*)d_in[1];  // [B,F]
  const float* sign = (const float*)d_in[2];  // [S,F]
  const float* ew   = (const float*)d_in[3];  // [E]
  const int*   ed   = (const int*)d_in[4];    // [E]
  const int*   es   = (const int*)d_in[5];    // [E]

  const int Bn = 64;
  const int F  = in_sizes[0] / Bn;            // 512
  const int S  = in_sizes[2] / F;             // 8192
  const int E  = in_sizes[3];                 // 500000
  const int D  = out_size / Bn;               // 1024

  // Workspace layout (floats): simT[S*Bn] | agg[D*Bn] | hinv[Bn] | gnorm[Bn] | sinv[S]
  float* ws    = (float*)d_ws;
  float* simT  = ws;
  float* agg   = simT + (size_t)S * Bn;
  float* hinv  = agg + (size_t)D * Bn;
  float* gnorm = hinv + Bn;
  float* sinv  = gnorm + Bn;

  // 1) row norms
  rownorm_kernel<<<Bn, 256, 0, stream>>>(heat, F, hinv, 1);
  rownorm_kernel<<<Bn, 256, 0, stream>>>(grad, F, gnorm, 0);
  rownorm_kernel<<<S, 256, 0, stream>>>(sign, F, sinv, 1);

  // 2) WMMA cosine-similarity GEMM -> simT[S][B]
  const int tiles = (S >> 4) * (Bn >> 4);     // 2048 wave-tiles
  simgemm_kernel<<<(tiles + 7) / 8, 256, 0, stream>>>(heat, sign, hinv, sinv,
                                                      simT, F, Bn, S);

  // 3) zero accumulator, edge scatter, finalize
  zero_kernel<<<(D * Bn + 255) / 256, 256, 0, stream>>>(agg, D * Bn);
  edge_scatter_kernel<<<512, 256, 0, stream>>>(simT, ew, ed, es, agg, E);
  const int total = Bn * D;
  finalize_kernel<<<(total + 255) / 256, 256, 0, stream>>>(agg, gnorm,
                                                           (float*)d_out, D, total);
}